// GNNDecoder_21251498180834
// MI455X (gfx1250) — compile-verified
//
#include <hip/hip_runtime.h>
#include <hip/hip_bf16.h>

typedef __attribute__((ext_vector_type(2))) float v2f;
typedef __attribute__((ext_vector_type(8))) float v8f;

#define BGRAPHS   2048
#define EMB       16
#define NNODES    89
#define H0        32
#define H1        64
#define OUTF      3
#define NTOT      182272LL      /* B * 89            */
#define ETOT      2916352LL     /* 16 * N            */
#define MTOT      (ETOT + NTOT) /* messages incl. self-loops */
#define XCOLS     2848          /* 89 * 32           */
#define BN_EPS    1e-5f

// ---------------------------------------------------------------------------
// K0: init. agg1 = 0, deg = 1 (self-loop weight), out = b2 broadcast.
// ---------------------------------------------------------------------------
__global__ void k_init(float* __restrict__ agg1, float* __restrict__ deg,
                       float* __restrict__ out, const float* __restrict__ b2) {
    long long i = (long long)blockIdx.x * blockDim.x + threadIdx.x;
    if (i < NTOT * H1) agg1[i] = 0.0f;
    if (i < NTOT)      deg[i]  = 1.0f;
    if (i < NTOT * OUTF) out[i] = b2[i % OUTF];
}

// ---------------------------------------------------------------------------
// K1: weighted in-degree: deg[dst] += w  (self-loop already in init)
// ---------------------------------------------------------------------------
__global__ void k_deg(const long long* __restrict__ dst,
                      const float* __restrict__ ew, float* __restrict__ deg) {
    long long e = (long long)blockIdx.x * blockDim.x + threadIdx.x;
    if (e >= ETOT) return;
    atomicAdd(&deg[dst[e]], ew[e]);
}

// ---------------------------------------------------------------------------
// K2: dinv = rsqrt(deg)  (deg >= 1 always) -- in place
// ---------------------------------------------------------------------------
__global__ void k_dinv(float* __restrict__ deg) {
    long long i = (long long)blockIdx.x * blockDim.x + threadIdx.x;
    if (i >= NTOT) return;
    deg[i] = rsqrtf(deg[i]);
}

// ---------------------------------------------------------------------------
// K2b: per-message norm = dinv[src] * w * dinv[dst]  (shared by both layers)
// ---------------------------------------------------------------------------
__global__ void k_norm(const long long* __restrict__ src,
                       const long long* __restrict__ dst,
                       const float* __restrict__ ew,
                       const float* __restrict__ dinv,
                       float* __restrict__ norm) {
    long long m = (long long)blockIdx.x * blockDim.x + threadIdx.x;
    if (m >= MTOT) return;
    long long s, d; float w;
    if (m < ETOT) { s = src[m]; d = dst[m]; w = ew[m]; }
    else          { s = d = m - ETOT;       w = 1.0f;  }
    norm[m] = dinv[s] * w * dinv[d];
}

// ---------------------------------------------------------------------------
// K3: expander GEMM via V_WMMA_F32_16X16X4_F32.
//     x[2048,2848] = ge[2048,16] @ Wexp[16,2848] + bexp   (== x[N,32])
//     One wave -> one 16x16 tile, K=16 -> 4 chained WMMAs.
// ---------------------------------------------------------------------------
__global__ void k_expand(const float* __restrict__ ge,
                         const float* __restrict__ Wexp,
                         const float* __restrict__ bexp,
                         float* __restrict__ x) {
    const int MT = BGRAPHS / 16;           // 128
    const int NT = XCOLS / 16;             // 178
    int wid = (int)((blockIdx.x * blockDim.x + threadIdx.x) >> 5);
    if (wid >= MT * NT) return;            // wave-uniform
    int mt = wid / NT, nt = wid % NT;
    int mbase = mt * 16, nbase = nt * 16;

    int lane = threadIdx.x & 31;
    int half = lane >> 4;                  // 0 | 1
    int lp   = lane & 15;

    v8f acc = {};
#pragma unroll
    for (int kb = 0; kb < EMB; kb += 4) {
        v2f a, b;
        int ka = kb + 2 * half;
        a.x = ge[(mbase + lp) * EMB + ka];
        a.y = ge[(mbase + lp) * EMB + ka + 1];
        b.x = Wexp[(ka)     * XCOLS + nbase + lp];
        b.y = Wexp[(ka + 1) * XCOLS + nbase + lp];
        acc = __builtin_amdgcn_wmma_f32_16x16x4_f32(
            false, a, false, b, (short)0, acc, false, false);
    }
    float bias = bexp[nbase + lp];
#pragma unroll
    for (int i = 0; i < 8; ++i)
        x[(long long)(mbase + i + 8 * half) * XCOLS + nbase + lp] = acc[i] + bias;
}

// ---------------------------------------------------------------------------
// K4: h1[N,64] = x[N,32] @ W1[32,64]  via V_WMMA_F32_16X16X4_F32 (K=32 -> 8)
// ---------------------------------------------------------------------------
__global__ void k_gemm1(const float* __restrict__ x,
                        const float* __restrict__ W1,
                        float* __restrict__ h1) {
    const int MT = (int)(NTOT / 16);       // 11392
    const int NT = H1 / 16;                // 4
    int wid = (int)((blockIdx.x * blockDim.x + threadIdx.x) >> 5);
    if (wid >= MT * NT) return;            // wave-uniform
    int mt = wid >> 2, nt = wid & 3;
    long long mbase = (long long)mt * 16;
    int nbase = nt * 16;

    int lane = threadIdx.x & 31;
    int half = lane >> 4;
    int lp   = lane & 15;

    v8f acc = {};
#pragma unroll
    for (int kb = 0; kb < H0; kb += 4) {
        v2f a, b;
        int ka = kb + 2 * half;
        a.x = x[(mbase + lp) * H0 + ka];
        a.y = x[(mbase + lp) * H0 + ka + 1];
        b.x = W1[(ka)     * H1 + nbase + lp];
        b.y = W1[(ka + 1) * H1 + nbase + lp];
        acc = __builtin_amdgcn_wmma_f32_16x16x4_f32(
            false, a, false, b, (short)0, acc, false, false);
    }
#pragma unroll
    for (int i = 0; i < 8; ++i)
        h1[(mbase + i + 8 * half) * H1 + nbase + lp] = acc[i];
}

// ---------------------------------------------------------------------------
// K5: layer-1 aggregation. 16 threads/message, float4 gather + 4 atomics.
//     h1 rows are 256B aligned -> b128 loads are legal & coalesced.
// ---------------------------------------------------------------------------
__global__ void k_scatter1(const long long* __restrict__ src,
                           const long long* __restrict__ dst,
                           const float* __restrict__ norm,
                           const float* __restrict__ h1,
                           float* __restrict__ agg1) {
    long long t = (long long)blockIdx.x * blockDim.x + threadIdx.x;
    long long m = t >> 4;
    int q = (int)(t & 15);                 // which float4 of the 64-wide row
    if (m >= MTOT) return;
    long long s, d;
    if (m < ETOT) { s = src[m]; d = dst[m]; }
    else          { s = d = m - ETOT;       }
    float nm = norm[m];
    const float4 v = ((const float4*)(h1 + s * H1))[q];
    float* ap = agg1 + d * H1 + q * 4;
    atomicAdd(ap + 0, nm * v.x);
    atomicAdd(ap + 1, nm * v.y);
    atomicAdd(ap + 2, nm * v.z);
    atomicAdd(ap + 3, nm * v.w);
}

// ---------------------------------------------------------------------------
// K6: +b1, BatchNorm(eval), ReLU -- in place on agg1
// ---------------------------------------------------------------------------
__global__ void k_bnrelu(float* __restrict__ agg1,
                         const float* __restrict__ b1,
                         const float* __restrict__ gamma,
                         const float* __restrict__ beta,
                         const float* __restrict__ mean,
                         const float* __restrict__ var) {
    long long i = (long long)blockIdx.x * blockDim.x + threadIdx.x;
    if (i >= NTOT * H1) return;
    int c = (int)(i & 63);
    float v = agg1[i] + b1[c];
    v = (v - mean[c]) * rsqrtf(var[c] + BN_EPS) * gamma[c] + beta[c];
    agg1[i] = v > 0.0f ? v : 0.0f;
}

// ---------------------------------------------------------------------------
// K7: h2[N,3] = xbn[N,64] @ W2[64,3]  (3 cols -> scalar FMA, float4 loads)
// ---------------------------------------------------------------------------
__global__ void k_gemm2(const float* __restrict__ xbn,
                        const float* __restrict__ W2,
                        float* __restrict__ h2) {
    long long n = (long long)blockIdx.x * blockDim.x + threadIdx.x;
    if (n >= NTOT) return;
    const float4* xr = (const float4*)(xbn + n * H1);
    float a0 = 0.f, a1 = 0.f, a2 = 0.f;
#pragma unroll
    for (int k4 = 0; k4 < H1 / 4; ++k4) {
        float4 v = xr[k4];
        int k = k4 * 4;
        a0 = fmaf(v.x, W2[(k+0)*OUTF+0], a0); a1 = fmaf(v.x, W2[(k+0)*OUTF+1], a1); a2 = fmaf(v.x, W2[(k+0)*OUTF+2], a2);
        a0 = fmaf(v.y, W2[(k+1)*OUTF+0], a0); a1 = fmaf(v.y, W2[(k+1)*OUTF+1], a1); a2 = fmaf(v.y, W2[(k+1)*OUTF+2], a2);
        a0 = fmaf(v.z, W2[(k+2)*OUTF+0], a0); a1 = fmaf(v.z, W2[(k+2)*OUTF+1], a1); a2 = fmaf(v.z, W2[(k+2)*OUTF+2], a2);
        a0 = fmaf(v.w, W2[(k+3)*OUTF+0], a0); a1 = fmaf(v.w, W2[(k+3)*OUTF+1], a1); a2 = fmaf(v.w, W2[(k+3)*OUTF+2], a2);
    }
    h2[n * OUTF + 0] = a0;
    h2[n * OUTF + 1] = a1;
    h2[n * OUTF + 2] = a2;
}

// ---------------------------------------------------------------------------
// K8: layer-2 aggregation into d_out (pre-initialized to b2). 4 thr/message.
// ---------------------------------------------------------------------------
__global__ void k_scatter2(const long long* __restrict__ src,
                           const long long* __restrict__ dst,
                           const float* __restrict__ norm,
                           const float* __restrict__ h2,
                           float* __restrict__ out) {
    long long t = (long long)blockIdx.x * blockDim.x + threadIdx.x;
    long long m = t >> 2;
    int f = (int)(t & 3);
    if (m >= MTOT || f >= OUTF) return;
    long long s, d;
    if (m < ETOT) { s = src[m]; d = dst[m]; }
    else          { s = d = m - ETOT;       }
    float nm = norm[m];
    atomicAdd(&out[d * OUTF + f], nm * h2[s * OUTF + f]);
}

// ---------------------------------------------------------------------------
extern "C" void kernel_launch(void* const* d_in, const int* in_sizes, int n_in,
                              void* d_out, int out_size, void* d_ws, size_t ws_size,
                              hipStream_t stream) {
    const float*     ge    = (const float*)d_in[0];
    const long long* eidx  = (const long long*)d_in[1];
    const float*     ew    = (const float*)d_in[2];
    const float*     Wexp  = (const float*)d_in[3];
    const float*     bexp  = (const float*)d_in[4];
    const float*     W1    = (const float*)d_in[5];
    const float*     b1    = (const float*)d_in[6];
    const float*     W2    = (const float*)d_in[7];
    const float*     b2    = (const float*)d_in[8];
    const float*     gam   = (const float*)d_in[9];
    const float*     bet   = (const float*)d_in[10];
    const float*     mean  = (const float*)d_in[11];
    const float*     var   = (const float*)d_in[12];
    float*           out   = (float*)d_out;

    const long long* src = eidx;
    const long long* dst = eidx + ETOT;

    // workspace layout (fp32, 256B-aligned regions)
    float* ws = (float*)d_ws;
    size_t off = 0;
    auto take = [&](size_t n) { float* p = ws + off; off += (n + 63) & ~(size_t)63; return p; };
    float* deg  = take(NTOT);            // becomes dinv after K2
    float* nrm  = take(MTOT);            // per-message norm
    float* x    = take(NTOT * H0);
    float* h1   = take(NTOT * H1);
    float* agg1 = take(NTOT * H1);       // becomes xbn after K6
    float* h2   = take(NTOT * OUTF);

    const int TB = 256;

    // K0: init
    {
        long long n = NTOT * H1;
        k_init<<<(unsigned)((n + TB - 1) / TB), TB, 0, stream>>>(agg1, deg, out, b2);
    }
    // K1: degree
    k_deg<<<(unsigned)((ETOT + TB - 1) / TB), TB, 0, stream>>>(dst, ew, deg);
    // K2: rsqrt
    k_dinv<<<(unsigned)((NTOT + TB - 1) / TB), TB, 0, stream>>>(deg);
    // K2b: per-message norm (shared by both layers)
    k_norm<<<(unsigned)((MTOT + TB - 1) / TB), TB, 0, stream>>>(src, dst, ew, deg, nrm);
    // K3: expander GEMM (WMMA f32) -- 128*178 = 22784 waves, 8 waves/block
    k_expand<<<(22784 + 7) / 8, TB, 0, stream>>>(ge, Wexp, bexp, x);
    // K4: h1 = x @ W1 (WMMA f32) -- 11392*4 = 45568 waves
    k_gemm1<<<(45568 + 7) / 8, TB, 0, stream>>>(x, W1, h1);
    // K5: aggregation layer 1 (16 threads/message, float4)
    {
        long long n = MTOT * 16;
        k_scatter1<<<(unsigned)((n + TB - 1) / TB), TB, 0, stream>>>(src, dst, nrm, h1, agg1);
    }
    // K6: bias + BN + ReLU
    {
        long long n = NTOT * H1;
        k_bnrelu<<<(unsigned)((n + TB - 1) / TB), TB, 0, stream>>>(agg1, b1, gam, bet, mean, var);
    }
    // K7: h2 = xbn @ W2
    k_gemm2<<<(unsigned)((NTOT + TB - 1) / TB), TB, 0, stream>>>(agg1, W2, h2);
    // K8: aggregation layer 2 into out
    {
        long long n = MTOT * 4;
        k_scatter2<<<(unsigned)((n + TB - 1) / TB), TB, 0, stream>>>(src, dst, nrm, h2, out);
    }
}